// InfoNCE_7645041786967
// MI455X (gfx1250) — compile-verified
//
#include <hip/hip_runtime.h>
#include <hip/hip_bf16.h>
#include <math.h>

typedef __attribute__((ext_vector_type(2))) float v2f;
typedef __attribute__((ext_vector_type(4))) float v4f;
typedef __attribute__((ext_vector_type(8))) float v8f;

#define BSZ 8192
#define DDIM 64
#define SEG 8
#define TILES_N 512               // 8192 / 16
#define TILES_N_PER_SEG (TILES_N / SEG)
#define STRIPS 256                // 8192 / 32 (32 rows per wave)
#define INV_TEMP (1.0f / 0.07f)

// -------- Kernel 1: reciprocal row norms (clamped to eps) --------
__global__ __launch_bounds__(256)
void infonce_norms(const float* __restrict__ fa, const float* __restrict__ fb,
                   float* __restrict__ ina, float* __restrict__ inb) {
    int row = blockIdx.x * 256 + threadIdx.x;
    if (row >= 2 * BSZ) return;
    const float* src = (row < BSZ) ? (fa + (size_t)row * DDIM)
                                   : (fb + (size_t)(row - BSZ) * DDIM);
    const v4f* p = (const v4f*)src;
    float s = 0.0f;
#pragma unroll
    for (int i = 0; i < DDIM / 4; ++i) {
        v4f v = p[i];
        s += v.x * v.x + v.y * v.y + v.z * v.z + v.w * v.w;
    }
    float inv = 1.0f / fmaxf(sqrtf(s), 1e-8f);
    if (row < BSZ) ina[row] = inv; else inb[row - BSZ] = inv;
}

// -- Kernel 2: WMMA f32 GEMM (32-row strip/wave) + cos write + row sums --
// Each wave: 2 M-tiles sharing every B fragment (32 WMMAs per 16 loads),
// num/den accumulated in registers across the segment, no atomics,
// non-temporal streaming stores for the 256MB cos_score output.
__global__ __launch_bounds__(256)
void infonce_gemm(const float* __restrict__ fa, const float* __restrict__ fb,
                  const int* __restrict__ labels,
                  const float* __restrict__ ina, const float* __restrict__ inb,
                  float* __restrict__ cos_out,
                  float* __restrict__ numP, float* __restrict__ denP) {
    const int lane   = threadIdx.x & 31;
    const int wave   = threadIdx.x >> 5;
    const int strip  = blockIdx.x * 8 + wave;   // 0..255 (32 rows each)
    const int seg    = blockIdx.y;              // 0..SEG-1
    const int half   = lane >> 4;               // C/D: lanes16-31 hold rows +8
    const int l15    = lane & 15;
    const int row0   = strip * 32;

    // A fragments for both 16-row M-tiles (16x4 f32 layout).
    v2f afrag[2][16];
#pragma unroll
    for (int mt = 0; mt < 2; ++mt) {
        const float* arow = fa + (size_t)(row0 + mt * 16 + l15) * DDIM + 2 * half;
#pragma unroll
        for (int s = 0; s < 16; ++s)
            afrag[mt][s] = *(const v2f*)(arow + 4 * s);
    }

    // Per-lane row constants: C/D VGPR r maps to row = base + r + 8*half.
    float inva[2][8];
    int   labm[2][8];
#pragma unroll
    for (int mt = 0; mt < 2; ++mt)
#pragma unroll
        for (int r = 0; r < 8; ++r) {
            int row      = row0 + mt * 16 + r + 8 * half;
            inva[mt][r]  = ina[row];
            labm[mt][r]  = labels[row];
        }

    float num_acc[2][8] = {};
    float den_acc[2][8] = {};

    const int tn0 = seg * TILES_N_PER_SEG;
    for (int tn = tn0; tn < tn0 + TILES_N_PER_SEG; ++tn) {
        const int n = tn * 16 + l15;
        const float* brow = fb + (size_t)n * DDIM + 2 * half;  // B^T fragment
        v8f acc0 = {};
        v8f acc1 = {};
#pragma unroll
        for (int s = 0; s < 16; ++s) {
            v2f bf = *(const v2f*)(brow + 4 * s);
            acc0 = __builtin_amdgcn_wmma_f32_16x16x4_f32(
                false, afrag[0][s], false, bf, (short)0, acc0, false, false);
            acc1 = __builtin_amdgcn_wmma_f32_16x16x4_f32(
                false, afrag[1][s], false, bf, (short)0, acc1, false, false);
        }
        const float invb = inb[n];
        const int   labn = labels[n];
#pragma unroll
        for (int mt = 0; mt < 2; ++mt) {
#pragma unroll
            for (int r = 0; r < 8; ++r) {
                float  c   = (mt ? acc1[r] : acc0[r]) * inva[mt][r] * invb;
                size_t row = (size_t)(row0 + mt * 16 + r + 8 * half);
                __builtin_nontemporal_store(c, cos_out + row * BSZ + n);
                float e = __expf(c * INV_TEMP);
                den_acc[mt][r] += e;
                num_acc[mt][r] += (labm[mt][r] == labn) ? e : 0.0f;
            }
        }
    }

    // Butterfly reduce within each 16-lane half; exclusive store of partials.
#pragma unroll
    for (int mt = 0; mt < 2; ++mt)
#pragma unroll
        for (int r = 0; r < 8; ++r) {
            float dsum = den_acc[mt][r], nsum = num_acc[mt][r];
#pragma unroll
            for (int m = 1; m < 16; m <<= 1) {
                dsum += __shfl_xor(dsum, m, 32);
                nsum += __shfl_xor(nsum, m, 32);
            }
            if (l15 == 0) {
                int row = row0 + mt * 16 + r + 8 * half;
                denP[(size_t)seg * BSZ + row] = dsum;
                numP[(size_t)seg * BSZ + row] = nsum;
            }
        }
}

// ---------------- Kernel 3: final loss reduction ----------------
__global__ __launch_bounds__(256)
void infonce_loss(const float* __restrict__ numP, const float* __restrict__ denP,
                  float* __restrict__ out) {
    __shared__ float red[256];
    float local = 0.0f;
    for (int i = threadIdx.x; i < BSZ; i += 256) {
        float num = 0.0f, den = 0.0f;
#pragma unroll
        for (int s = 0; s < SEG; ++s) {
            num += numP[(size_t)s * BSZ + i];
            den += denP[(size_t)s * BSZ + i];
        }
        if (num == 0.0f) num = 0.01f;
        local += logf(num / den);
    }
    red[threadIdx.x] = local;
    __syncthreads();
    for (int s = 128; s > 0; s >>= 1) {
        if (threadIdx.x < s) red[threadIdx.x] += red[threadIdx.x + s];
        __syncthreads();
    }
    if (threadIdx.x == 0) out[0] = -red[0] / (float)BSZ;
}

extern "C" void kernel_launch(void* const* d_in, const int* in_sizes, int n_in,
                              void* d_out, int out_size, void* d_ws, size_t ws_size,
                              hipStream_t stream) {
    const int*   labels = (const int*)d_in[0];    // label_tensor [8192]
    const float* fa     = (const float*)d_in[1];  // feature_a [8192,64]
    const float* fb     = (const float*)d_in[2];  // feature_b [8192,64]
    float* out = (float*)d_out;                   // [0]=loss, [1..]=cos_score

    float* ws   = (float*)d_ws;
    float* ina  = ws;                     // 8192 reciprocal norms of A rows
    float* inb  = ws + BSZ;               // 8192 reciprocal norms of B rows
    float* numP = ws + 2 * BSZ;           // SEG*8192 partial numerators
    float* denP = ws + 2 * BSZ + (size_t)SEG * BSZ;  // SEG*8192 partial denoms
    float* cos_out = out + 1;

    infonce_norms<<<(2 * BSZ + 255) / 256, 256, 0, stream>>>(fa, fb, ina, inb);
    dim3 grid(STRIPS / 8, SEG);           // 32 blocks x 8 waves = 256 strips, x8 segs
    infonce_gemm<<<grid, 256, 0, stream>>>(fa, fb, labels, ina, inb, cos_out, numP, denP);
    infonce_loss<<<1, 256, 0, stream>>>(numP, denP, out);
}